// VariationalQuantumCircuit_56367150793024
// MI455X (gfx1250) — compile-verified
//
#include <hip/hip_runtime.h>
#include <hip/hip_bf16.h>

// ---------------------------------------------------------------------------
// Variational quantum circuit simulator, CDNA5 (gfx1250) WMMA formulation.
//
// State per sample: 32x32 complex matrix S (rows = wires 0-4, cols = wires 5-9;
// wire i <-> amplitude bit (9-i), matching the reference flatten order).
// Per layer:
//   GEMM1: S <- S * U_low^T      (RZ*RY on wires 5-9)                [128 wmma]
//   GEMM2: S <- U_high * S       (RZ*RY on wires 0-4 + CNOT chain)   [128 wmma]
//   remaining CNOTs = closed-form index permutation, folded into the next
//   layer's (or the measurement's) read addressing -> zero cost.
// Encoding = real outer product rowv (x) colv. Measurement = signed |amp|^2
// sums + wave32 butterfly reduction.
// One wave32 per sample, state in LDS (2 buffers x re/im x 32x33 f32 ~ 16.9KB).
// ---------------------------------------------------------------------------

typedef __attribute__((ext_vector_type(2))) float v2f;
typedef __attribute__((ext_vector_type(4))) float v4f;
typedef __attribute__((ext_vector_type(8))) float v8f;

#define NQ      10
#define NL      4
#define BATCH_N 8192
#define PITCH   33                 // 32 floats + 1 pad -> conflict-free banks
#define PLANE   (32 * PITCH)
#define SAMPLE_F (4 * PLANE)       // Sa_re, Sa_im, Sb_re, Sb_im
#define WPB     2                  // waves (=samples) per block

// ws layout (floats): UH[l][i][k][p][256] then UL[l][k][j][p][256]
// fragments stored lane-major: frag_base + lane*8 + slot  (slot = 2*kk + vg)
__device__ __forceinline__ int uh_off(int l, int i, int k, int p) {
  return (((l * 2 + i) * 2 + k) * 2 + p) * 256;
}
__device__ __forceinline__ int ul_off(int l, int k, int j, int p) {
  return 8192 + (((l * 2 + k) * 2 + j) * 2 + p) * 256;
}

// G = RZ(b) * RY(a); element (rb, cb) of the 2x2 complex gate.
__device__ __forceinline__ void gate_elem(float a, float b, int rb, int cb,
                                          float& fr, float& fi) {
  float ca = cosf(0.5f * a), sa = sinf(0.5f * a);
  float cz = cosf(0.5f * b), sz = sinf(0.5f * b);
  float ry = rb ? (cb ? ca : sa) : (cb ? -sa : ca);
  fr = ry * cz;
  fi = ry * (rb ? sz : -sz);   // row0 phase e^{-ib/2}, row1 phase e^{+ib/2}
}

// ---------------------------------------------------------------------------
// Prep: build per-layer 32x32 group unitaries directly in WMMA operand layout.
//   UH = (CNOT chain on rows) * tensor(G_0..G_4)   -> A-layout fragments
//   UL^T fragments of tensor(G_5..G_9)             -> B-layout fragments
// A/B slot map (f32 16x16x4): M|N = lane&15, K = 4*(slot>>1)+(slot&1)+2*(lane>>4)
// ---------------------------------------------------------------------------
__global__ void vqc_prep(const float* __restrict__ w, float* __restrict__ ws) {
  int l = blockIdx.x;
  for (int t = threadIdx.x; t < 4096; t += blockDim.x) {
    int which = t >> 11;       // 0 = UH, 1 = UL
    int e     = t & 2047;
    int frag  = e >> 8;
    int lane  = (e >> 3) & 31;
    int slot  = e & 7;
    int p     = frag & 1;
    int Kloc  = 4 * (slot >> 1) + (slot & 1) + 2 * (lane >> 4);

    float vr = 1.f, vi = 0.f;
    int off;
    if (which == 0) {
      int it = frag >> 2, kt = (frag >> 1) & 1;
      int m = it * 16 + (lane & 15);          // output row (post-CNOT-chain)
      int kcol = kt * 16 + Kloc;              // contraction col
      int r0 = (m ^ (m >> 1)) & 31;           // inverse of CNOT-chain row perm
      #pragma unroll
      for (int w2 = 0; w2 < 5; ++w2) {
        int rb = (r0 >> (4 - w2)) & 1;
        int cb = (kcol >> (4 - w2)) & 1;
        float fr, fi;
        gate_elem(w[(l * NQ + w2) * 2], w[(l * NQ + w2) * 2 + 1], rb, cb, fr, fi);
        float nr = vr * fr - vi * fi; vi = vr * fi + vi * fr; vr = nr;
      }
      off = uh_off(l, it, kt, p);
    } else {
      int kt = frag >> 2, jt = (frag >> 1) & 1;
      int n  = jt * 16 + (lane & 15);         // B col index (output col of S')
      int kg = kt * 16 + Kloc;                // contraction index
      #pragma unroll
      for (int w2 = 0; w2 < 5; ++w2) {
        int rb = (n >> (4 - w2)) & 1;         // U_low^T[k][n] = U_low[n][k]
        int cb = (kg >> (4 - w2)) & 1;
        float fr, fi;
        gate_elem(w[(l * NQ + 5 + w2) * 2], w[(l * NQ + 5 + w2) * 2 + 1], rb, cb, fr, fi);
        float nr = vr * fr - vi * fi; vi = vr * fi + vi * fr; vr = nr;
      }
      off = ul_off(l, kt, jt, p);
    }
    ws[off + lane * 8 + slot] = p ? vi : vr;
  }
}

// ---------------------------------------------------------------------------
// Main: one wave32 per sample.
// ---------------------------------------------------------------------------
__global__ __launch_bounds__(32 * WPB) void vqc_main(const float* __restrict__ x,
                                                     const float* __restrict__ ws,
                                                     float* __restrict__ out) {
  __shared__ float lds[WPB * SAMPLE_F];
  const int lane = threadIdx.x & 31;
  const int wv   = threadIdx.x >> 5;
  const int sample = blockIdx.x * WPB + wv;

  float* Sa_r = lds + wv * SAMPLE_F;
  float* Sa_i = Sa_r + PLANE;
  float* Sb_r = Sa_r + 2 * PLANE;
  float* Sb_i = Sa_r + 3 * PLANE;

  // ---- angle encoding: product state, amp(r,c) = rowv[r] * colv[c] (real) ----
  float rowv = 1.f, colv = 1.f;
  #pragma unroll
  for (int w2 = 0; w2 < 5; ++w2) {
    float ah = 0.5f * x[sample * NQ + w2];
    float al = 0.5f * x[sample * NQ + 5 + w2];
    int b = (lane >> (4 - w2)) & 1;
    rowv *= b ? sinf(ah) : cosf(ah);
    colv *= b ? sinf(al) : cosf(al);
  }
  for (int r = 0; r < 32; ++r) {
    float rv = __shfl(rowv, r, 32);
    Sa_r[r * PITCH + lane] = rv * colv;
    Sa_i[r * PITCH + lane] = 0.f;
  }

  for (int l = 0; l < NL; ++l) {
    // ---------------- GEMM1: Sb = Perm^{-1}(Sa) * U_low^T ----------------
    #pragma unroll
    for (int it = 0; it < 2; ++it) {
      float ar[2][8], ai[2][8];
      #pragma unroll
      for (int k = 0; k < 2; ++k)
        #pragma unroll
        for (int s = 0; s < 8; ++s) {
          int row = it * 16 + (lane & 15);
          int col = k * 16 + 4 * (s >> 1) + (s & 1) + 2 * (lane >> 4);
          int rs = row, cs = col;
          if (l > 0) {  // fold previous layer's CNOT permutation (inverse map)
            int rm = (col & 1) ? (row ^ 16) : row;     // undo CNOT(9,0)
            int c3 = (col ^ (col >> 1)) & 31;          // undo low CNOT chain
            cs = (rm & 1) ? (c3 ^ 16) : c3;            // undo CNOT(4,5)
            rs = rm;
          }
          ar[k][s] = Sa_r[rs * PITCH + cs];
          ai[k][s] = Sa_i[rs * PITCH + cs];
        }
      #pragma unroll
      for (int j = 0; j < 2; ++j) {
        v8f dr = {0,0,0,0,0,0,0,0}, di = {0,0,0,0,0,0,0,0};
        #pragma unroll
        for (int k = 0; k < 2; ++k) {
          const v4f* pr = (const v4f*)(ws + ul_off(l, k, j, 0) + lane * 8);
          const v4f* pi = (const v4f*)(ws + ul_off(l, k, j, 1) + lane * 8);
          v4f r0 = pr[0], r1 = pr[1], i0 = pi[0], i1 = pi[1];
          float br[8] = {r0.x, r0.y, r0.z, r0.w, r1.x, r1.y, r1.z, r1.w};
          float bi[8] = {i0.x, i0.y, i0.z, i0.w, i1.x, i1.y, i1.z, i1.w};
          #pragma unroll
          for (int kk = 0; kk < 4; ++kk) {
            v2f aR  = { ar[k][2 * kk],  ar[k][2 * kk + 1]};
            v2f aI  = { ai[k][2 * kk],  ai[k][2 * kk + 1]};
            v2f aIn = {-ai[k][2 * kk], -ai[k][2 * kk + 1]};
            v2f bR  = { br[2 * kk],     br[2 * kk + 1]};
            v2f bI  = { bi[2 * kk],     bi[2 * kk + 1]};
            dr = __builtin_amdgcn_wmma_f32_16x16x4_f32(false, aR,  false, bR, (short)0, dr, false, false);
            dr = __builtin_amdgcn_wmma_f32_16x16x4_f32(false, aIn, false, bI, (short)0, dr, false, false);
            di = __builtin_amdgcn_wmma_f32_16x16x4_f32(false, aR,  false, bI, (short)0, di, false, false);
            di = __builtin_amdgcn_wmma_f32_16x16x4_f32(false, aI,  false, bR, (short)0, di, false, false);
          }
        }
        #pragma unroll
        for (int s = 0; s < 8; ++s) {
          int m = it * 16 + s + 8 * (lane >> 4);
          int n = j * 16 + (lane & 15);
          Sb_r[m * PITCH + n] = dr[s];
          Sb_i[m * PITCH + n] = di[s];
        }
      }
    }
    // ---------------- GEMM2: Sa = U_high * Sb ----------------
    #pragma unroll
    for (int j = 0; j < 2; ++j) {
      float br[2][8], bi[2][8];
      #pragma unroll
      for (int k = 0; k < 2; ++k)
        #pragma unroll
        for (int s = 0; s < 8; ++s) {
          int K = k * 16 + 4 * (s >> 1) + (s & 1) + 2 * (lane >> 4);
          int n = j * 16 + (lane & 15);
          br[k][s] = Sb_r[K * PITCH + n];
          bi[k][s] = Sb_i[K * PITCH + n];
        }
      #pragma unroll
      for (int it = 0; it < 2; ++it) {
        v8f dr = {0,0,0,0,0,0,0,0}, di = {0,0,0,0,0,0,0,0};
        #pragma unroll
        for (int k = 0; k < 2; ++k) {
          const v4f* pr = (const v4f*)(ws + uh_off(l, it, k, 0) + lane * 8);
          const v4f* pi = (const v4f*)(ws + uh_off(l, it, k, 1) + lane * 8);
          v4f r0 = pr[0], r1 = pr[1], i0 = pi[0], i1 = pi[1];
          float ur[8] = {r0.x, r0.y, r0.z, r0.w, r1.x, r1.y, r1.z, r1.w};
          float ui[8] = {i0.x, i0.y, i0.z, i0.w, i1.x, i1.y, i1.z, i1.w};
          #pragma unroll
          for (int kk = 0; kk < 4; ++kk) {
            v2f aR  = { ur[2 * kk],  ur[2 * kk + 1]};
            v2f aI  = { ui[2 * kk],  ui[2 * kk + 1]};
            v2f aIn = {-ui[2 * kk], -ui[2 * kk + 1]};
            v2f bR  = { br[k][2 * kk], br[k][2 * kk + 1]};
            v2f bI  = { bi[k][2 * kk], bi[k][2 * kk + 1]};
            dr = __builtin_amdgcn_wmma_f32_16x16x4_f32(false, aR,  false, bR, (short)0, dr, false, false);
            dr = __builtin_amdgcn_wmma_f32_16x16x4_f32(false, aIn, false, bI, (short)0, dr, false, false);
            di = __builtin_amdgcn_wmma_f32_16x16x4_f32(false, aR,  false, bI, (short)0, di, false, false);
            di = __builtin_amdgcn_wmma_f32_16x16x4_f32(false, aI,  false, bR, (short)0, di, false, false);
          }
        }
        #pragma unroll
        for (int s = 0; s < 8; ++s) {
          int m = it * 16 + s + 8 * (lane >> 4);
          int n = j * 16 + (lane & 15);
          Sa_r[m * PITCH + n] = dr[s];
          Sa_i[m * PITCH + n] = di[s];
        }
      }
    }
  }

  // ---- measurement: apply pending CNOT perm at read time, signed |amp|^2 ----
  float acc[NQ];
  #pragma unroll
  for (int i = 0; i < NQ; ++i) acc[i] = 0.f;
  for (int r = 0; r < 32; ++r) {
    float re = Sa_r[r * PITCH + lane];
    float im = Sa_i[r * PITCH + lane];
    float p  = re * re + im * im;
    int c3 = (r & 1) ? (lane ^ 16) : lane;                         // CNOT(4,5)
    int c4 = (c3 ^ (c3 >> 1) ^ (c3 >> 2) ^ (c3 >> 3) ^ (c3 >> 4)) & 31;  // low chain
    int r2 = (c4 & 1) ? (r ^ 16) : r;                              // CNOT(9,0)
    int a  = (r2 << 5) | c4;                                       // logical index
    #pragma unroll
    for (int i = 0; i < NQ; ++i)
      acc[i] += (a & (1 << (9 - i))) ? -p : p;
  }
  #pragma unroll
  for (int off = 16; off > 0; off >>= 1)
    #pragma unroll
    for (int i = 0; i < NQ; ++i)
      acc[i] += __shfl_xor(acc[i], off, 32);
  if (lane == 0) {
    #pragma unroll
    for (int i = 0; i < NQ; ++i) out[sample * NQ + i] = acc[i];
  }
}

extern "C" void kernel_launch(void* const* d_in, const int* in_sizes, int n_in,
                              void* d_out, int out_size, void* d_ws, size_t ws_size,
                              hipStream_t stream) {
  const float* x = (const float*)d_in[0];     // (8192, 10) f32
  const float* w = (const float*)d_in[1];     // (4, 10, 2)  f32
  float* ws  = (float*)d_ws;                  // needs 16384 floats (64 KB)
  float* out = (float*)d_out;                 // (8192, 10) f32

  vqc_prep<<<NL, 1024, 0, stream>>>(w, ws);
  vqc_main<<<BATCH_N / WPB, 32 * WPB, 0, stream>>>(x, ws, out);
}